// PAN_20306605375903
// MI455X (gfx1250) — compile-verified
//
#include <hip/hip_runtime.h>

// 3D neighborhood attention (kernel_size=3) for q,k: [1,48,48,48,48], nh=8, hd=6.
// Memory-bound (~53 MB total traffic -> ~2.3us at 23.3 TB/s HBM). QK dots are
// exact fp32 via V_WMMA_F32_16X16X4_F32 (two chained K=4 WMMAs, head_dim 6
// padded to 8). k/q tiles staged to LDS with GLOBAL_LOAD_ASYNC_TO_LDS_B128
// (ASYNCcnt) when available, else plain float4 loads.

#define NH 8
#define HD 6
#define NT 27
#define N3 (48*48*48)
#define KSTRIDE 52      // padded C stride in LDS (multiple of 4 -> 16B rows for b128)
#define ATTN_STRIDE 28  // padded 27

typedef float v2f __attribute__((ext_vector_type(2)));
typedef float v8f __attribute__((ext_vector_type(8)));
typedef int   v4i __attribute__((vector_size(16)));

#if __has_builtin(__builtin_amdgcn_global_load_async_to_lds_b128)
#define USE_ASYNC_LDS 1
typedef __attribute__((address_space(1))) v4i* g_v4i_ptr;
typedef __attribute__((address_space(3))) v4i* l_v4i_ptr;
#else
#define USE_ASYNC_LDS 0
#endif

__device__ __forceinline__ void stage16(const float* __restrict__ gsrc, float* ldst) {
#if USE_ASYNC_LDS
    __builtin_amdgcn_global_load_async_to_lds_b128((g_v4i_ptr)gsrc, (l_v4i_ptr)ldst, 0, 0);
#else
    *(float4*)ldst = *(const float4*)gsrc;
#endif
}

__device__ __forceinline__ void wait_async_stage() {
#if USE_ASYNC_LDS
#if __has_builtin(__builtin_amdgcn_s_wait_asynccnt)
    __builtin_amdgcn_s_wait_asynccnt(0);
#else
    asm volatile("s_wait_asynccnt 0" ::: "memory");
#endif
#endif
}

__global__ __launch_bounds__(128)
void natten3d_wmma_kernel(const float* __restrict__ q,
                          const float* __restrict__ k,
                          const float* __restrict__ rpb,
                          float* __restrict__ out)
{
    __shared__ __align__(16) float kbuf[9 * 18 * KSTRIDE];  // 9 dirs x 18 z x C (raw)
    __shared__ __align__(16) float qbuf[16 * KSTRIDE];      // 16 z x C (raw)
    __shared__ float Sbuf[4 * 256];                          // per-wave 16x16 D tile
    __shared__ float attnbuf[NH * 16 * ATTN_STRIDE];         // logits per (head, m)
    __shared__ float rpbuf[NH * NT];

    const int x  = blockIdx.x;
    const int y  = blockIdx.y;
    const int z0 = blockIdx.z * 16;
    const int tid  = threadIdx.x;
    const int lane = tid & 31;
    const int wid  = tid >> 5;
    const int half = lane >> 4;    // lane-half selects K (A/B) or M+8 (D)
    const int mrow = lane & 15;
    const float scale = 0.40824829046386301637f;  // 6^-0.5, folded into the logit

    // ---- Stage k tile: 9 neighbor (x,y) columns, z in [z0-1, z0+16], zero-padded ----
    for (int idx = tid; idx < 9 * 18 * 12; idx += 128) {
        int dir = idx / (18 * 12);
        int rem = idx - dir * (18 * 12);
        int r   = rem / 12;
        int cw  = rem - r * 12;
        int xx = x + dir / 3 - 1;
        int yy = y + dir % 3 - 1;
        int zz = z0 - 1 + r;
        float* ldst = &kbuf[(dir * 18 + r) * KSTRIDE + cw * 4];
        if ((unsigned)xx < 48u && (unsigned)yy < 48u && (unsigned)zz < 48u) {
            stage16(k + (((size_t)(xx * 48 + yy) * 48 + zz) * 48 + cw * 4), ldst);
        } else {
            float4 zv = {0.f, 0.f, 0.f, 0.f};
            *(float4*)ldst = zv;
        }
    }
    // ---- Stage q tile (raw; scale applied to logits later) ----
    for (int idx = tid; idx < 16 * 12; idx += 128) {
        int r = idx / 12, cw = idx - r * 12;
        stage16(q + (((size_t)(x * 48 + y) * 48 + (z0 + r)) * 48 + cw * 4),
                &qbuf[r * KSTRIDE + cw * 4]);
    }
    // ---- Stage rpb ----
    for (int idx = tid; idx < NH * NT; idx += 128) rpbuf[idx] = rpb[idx];
    wait_async_stage();
    __syncthreads();

    // ---- Each wave: 2 heads x 9 directions, QK via WMMA f32 16x16x4 (exact fp32) ----
    for (int hh = 0; hh < 2; ++hh) {
        const int n = wid * 2 + hh;
        const float* qrow = &qbuf[mrow * KSTRIDE + n * HD];
        // A (16x4 f32): VGPR j holds K = half*2 + j (chunk0), 4 + half*2 + j (chunk1).
        v2f a0, a1;
        a0.x = qrow[half * 2 + 0];
        a0.y = qrow[half * 2 + 1];
        a1.x = (half == 0) ? qrow[4] : 0.f;   // K=4 valid, K=6 padded
        a1.y = (half == 0) ? qrow[5] : 0.f;   // K=5 valid, K=7 padded
        for (int dir = 0; dir < 9; ++dir) {
            // B (4x16 f32): N = lane&15 -> k row zz = z0-1+N; same K split as A.
            const float* krow = &kbuf[(dir * 18 + mrow) * KSTRIDE + n * HD];
            v2f b0, b1;
            b0.x = krow[half * 2 + 0];
            b0.y = krow[half * 2 + 1];
            b1.x = (half == 0) ? krow[4] : 0.f;
            b1.y = (half == 0) ? krow[5] : 0.f;

            v8f acc = {};
            acc = __builtin_amdgcn_wmma_f32_16x16x4_f32(false, a0, false, b0,
                                                        (short)0, acc, false, false);
            acc = __builtin_amdgcn_wmma_f32_16x16x4_f32(false, a1, false, b1,
                                                        (short)0, acc, false, false);

            // Spill D tile (row = r + 8*half, col = mrow) for diagonal-band gather.
#pragma unroll
            for (int r = 0; r < 8; ++r)
                Sbuf[wid * 256 + (r + half * 8) * 16 + mrow] = acc[r];
            __builtin_amdgcn_wave_barrier();

            // attn[m, dl] = scale * S[m, m+dl+1] + rpb; j in [0,17], j>=16 via 6-MAC dot.
            // Static lane split: half 0 -> dlidx {0,1}; half 1 -> dlidx {2}.
            {
                const int m = mrow;
                const float* Sw = &Sbuf[wid * 256 + m * 16];
                float* arow = &attnbuf[(n * 16 + m) * ATTN_STRIDE + dir * 3];
                const float* rrow = &rpbuf[n * NT + dir * 3];
                if (half == 0) {
                    float v0 = Sw[m];                       // j = m, always in-tile
                    arow[0] = v0 * scale + rrow[0];
                    float v1;
                    if (m < 15) {
                        v1 = Sw[m + 1];                     // j = m+1
                    } else {
                        v1 = 0.f;                           // j = 16 patch
                        const float* qr = &qbuf[m * KSTRIDE + n * HD];
                        const float* kr = &kbuf[(dir * 18 + 16) * KSTRIDE + n * HD];
#pragma unroll
                        for (int d = 0; d < HD; ++d) v1 += qr[d] * kr[d];
                    }
                    arow[1] = v1 * scale + rrow[1];
                } else {
                    int j = m + 2;
                    float v2;
                    if (j < 16) {
                        v2 = Sw[j];
                    } else {
                        v2 = 0.f;                           // j = 16,17 patch
                        const float* qr = &qbuf[m * KSTRIDE + n * HD];
                        const float* kr = &kbuf[(dir * 18 + j) * KSTRIDE + n * HD];
#pragma unroll
                        for (int d = 0; d < HD; ++d) v2 += qr[d] * kr[d];
                    }
                    arow[2] = v2 * scale + rrow[2];
                }
            }
            __builtin_amdgcn_wave_barrier();
        }
    }
    __syncthreads();

    // ---- Softmax over 27 tokens + contraction with constant offsets (di,dj,dl) ----
    {
        const int n = (wid << 1) + half;
        const int m = mrow;
        const float* arow = &attnbuf[(n * 16 + m) * ATTN_STRIDE];
        float mx = arow[0];
#pragma unroll
        for (int t = 1; t < NT; ++t) mx = fmaxf(mx, arow[t]);
        float s = 0.f, sdi = 0.f, sdj = 0.f, sdl = 0.f;
#pragma unroll
        for (int t = 0; t < NT; ++t) {
            float e = __expf(arow[t] - mx);
            s   += e;
            sdi += e * (float)(t / 9 - 1);
            sdj += e * (float)((t / 3) % 3 - 1);
            sdl += e * (float)(t % 3 - 1);
        }
        float inv = 1.f / s;
        size_t base = ((size_t)x * 48 + y) * 48 + (size_t)(z0 + m);
        out[(size_t)(n * 3 + 0) * N3 + base] = sdi * inv;
        out[(size_t)(n * 3 + 1) * N3 + base] = sdj * inv;
        out[(size_t)(n * 3 + 2) * N3 + base] = sdl * inv;
    }
}

extern "C" void kernel_launch(void* const* d_in, const int* in_sizes, int n_in,
                              void* d_out, int out_size, void* d_ws, size_t ws_size,
                              hipStream_t stream) {
    (void)in_sizes; (void)n_in; (void)out_size; (void)d_ws; (void)ws_size;
    const float* q   = (const float*)d_in[0];
    const float* k   = (const float*)d_in[1];
    const float* rpb = (const float*)d_in[2];
    float* out = (float*)d_out;
    dim3 grid(48, 48, 3);
    dim3 block(128);
    natten3d_wmma_kernel<<<grid, block, 0, stream>>>(q, k, rpb, out);
}